// DecoderRNN_30623116820698
// MI455X (gfx1250) — compile-verified
//
#include <hip/hip_runtime.h>
#include <math.h>

typedef __attribute__((ext_vector_type(2))) float v2f;
typedef __attribute__((ext_vector_type(8))) float v8f;

#define H 2048
#define V 50257

// ---------------------------------------------------------------------------
// Kernel 1: x = relu(emb[input])   [H]
// ---------------------------------------------------------------------------
__global__ void k_embed_relu(const int* __restrict__ input,
                             const float* __restrict__ emb,
                             float* __restrict__ x) {
    int j = blockIdx.x * blockDim.x + threadIdx.x;
    if (j < H) {
        int idx = input[0];                 // vocab id (fits in 32 bits)
        float v = emb[(size_t)idx * H + j];
        x[j] = v > 0.0f ? v : 0.0f;
    }
}

// ---------------------------------------------------------------------------
// Kernel 2: gates[4H] = W_ih @ x + W_hh @ h + b_ih + b_hh
// One wave computes a 16-row strip via V_WMMA_F32_16X16X4_F32, K-step 4.
// A fragment: activation broadcast (all 16 M-rows identical).
//   lane L holds A[M=L%16][K=(L>>4)*2 + v] for VGPR v in {0,1}
// B fragment: lane L holds B[K=(L>>4)*2 + v][N=L%16] = W[n0+L%16][k0+(L>>4)*2+v]
// D: row 0 (== every row) lives in VGPR0 of lanes 0..15.
// ---------------------------------------------------------------------------
__global__ void __launch_bounds__(256) k_gates(
    const float* __restrict__ x, const float* __restrict__ h0,
    const float* __restrict__ Wih, const float* __restrict__ Whh,
    const float* __restrict__ bih, const float* __restrict__ bhh,
    float* __restrict__ gates) {
    __shared__ float sx[H];
    __shared__ float sh[H];
    const int tid = threadIdx.x;
    for (int i = tid; i < H; i += 256) { sx[i] = x[i]; sh[i] = h0[i]; }
    __syncthreads();

    const int wave  = tid >> 5;
    const int lane  = tid & 31;
    const int strip = blockIdx.x * 8 + wave;   // 0..511 (4H/16 strips)
    const int n0    = strip * 16;
    const int row   = n0 + (lane & 15);
    const int kLane = (lane >> 4) * 2;         // 0 or 2

    const float* pih = Wih + (size_t)row * H + kLane;
    const float* phh = Whh + (size_t)row * H + kLane;

    v8f acc_i = {};
    v8f acc_h = {};
    for (int k0 = 0; k0 < H; k0 += 4) {
        v2f a_x, a_h;
        a_x.x = sx[k0 + kLane]; a_x.y = sx[k0 + kLane + 1];
        a_h.x = sh[k0 + kLane]; a_h.y = sh[k0 + kLane + 1];
        v2f b_i = *(const v2f*)(pih + k0);
        v2f b_h = *(const v2f*)(phh + k0);
        __builtin_prefetch(pih + k0 + 256, 0, 1);
        __builtin_prefetch(phh + k0 + 256, 0, 1);
        acc_i = __builtin_amdgcn_wmma_f32_16x16x4_f32(
            false, a_x, false, b_i, (short)0, acc_i, false, false);
        acc_h = __builtin_amdgcn_wmma_f32_16x16x4_f32(
            false, a_h, false, b_h, (short)0, acc_h, false, false);
    }

    if (lane < 16) {
        int n = n0 + lane;
        gates[n] = acc_i[0] + acc_h[0] + bih[n] + bhh[n];
    }
}

// ---------------------------------------------------------------------------
// Kernel 3: elementwise LSTM cell update. Writes h_new/c_new into d_out.
// ---------------------------------------------------------------------------
__global__ void k_lstm(const float* __restrict__ gates,
                       const float* __restrict__ c0,
                       float* __restrict__ out) {
    int j = blockIdx.x * blockDim.x + threadIdx.x;
    if (j < H) {
        float ig = 1.0f / (1.0f + expf(-gates[j]));
        float fg = 1.0f / (1.0f + expf(-gates[H + j]));
        float gg = tanhf(gates[2 * H + j]);
        float og = 1.0f / (1.0f + expf(-gates[3 * H + j]));
        float c  = fg * c0[j] + ig * gg;
        float h  = og * tanhf(c);
        out[V + j]     = h;   // h_new
        out[V + H + j] = c;   // c_new
    }
}

// ---------------------------------------------------------------------------
// Kernel 4: logits[V] = W_out @ h_new + b_out   (same WMMA GEMV scheme)
// ---------------------------------------------------------------------------
__global__ void __launch_bounds__(256) k_logits(
    const float* __restrict__ hnew, const float* __restrict__ Wout,
    const float* __restrict__ bout, float* __restrict__ logits) {
    __shared__ float sh[H];
    const int tid = threadIdx.x;
    for (int i = tid; i < H; i += 256) sh[i] = hnew[i];
    __syncthreads();   // barrier before any wave may exit

    const int wave    = tid >> 5;
    const int lane    = tid & 31;
    const int strip   = blockIdx.x * 8 + wave;
    const int nStrips = (V + 15) / 16;         // 3142
    if (strip >= nStrips) return;              // wave-uniform, after barrier

    const int n0 = strip * 16;
    int row = n0 + (lane & 15);
    if (row >= V) row = V - 1;                 // clamp loads; store masked below
    const int kLane = (lane >> 4) * 2;

    const float* p = Wout + (size_t)row * H + kLane;

    v8f acc = {};
    for (int k0 = 0; k0 < H; k0 += 4) {
        v2f a;
        a.x = sh[k0 + kLane]; a.y = sh[k0 + kLane + 1];
        v2f b = *(const v2f*)(p + k0);
        __builtin_prefetch(p + k0 + 256, 0, 1);
        acc = __builtin_amdgcn_wmma_f32_16x16x4_f32(
            false, a, false, b, (short)0, acc, false, false);
    }

    if (lane < 16) {
        int n = n0 + lane;
        if (n < V) logits[n] = acc[0] + bout[n];
    }
}

// ---------------------------------------------------------------------------
// Kernel 5: in-place log-softmax over logits[V]. Single block of 1024 threads
// (32 wave32s): shuffle reduce within wave, LDS tree across waves.
// ---------------------------------------------------------------------------
__global__ void __launch_bounds__(1024) k_logsoftmax(float* __restrict__ logp) {
    __shared__ float red[32];
    const int tid  = threadIdx.x;
    const int lane = tid & 31;
    const int wave = tid >> 5;

    // pass 1: global max
    float m = -INFINITY;
    for (int i = tid; i < V; i += 1024) m = fmaxf(m, logp[i]);
    for (int off = 16; off > 0; off >>= 1) m = fmaxf(m, __shfl_xor(m, off, 32));
    if (lane == 0) red[wave] = m;
    __syncthreads();
    if (tid < 32) {
        float v = red[tid];
        for (int off = 16; off > 0; off >>= 1) v = fmaxf(v, __shfl_xor(v, off, 32));
        if (tid == 0) red[0] = v;
    }
    __syncthreads();
    const float gmax = red[0];
    __syncthreads();   // red[] reused below

    // pass 2: sum of exp
    float s = 0.0f;
    for (int i = tid; i < V; i += 1024) s += expf(logp[i] - gmax);
    for (int off = 16; off > 0; off >>= 1) s += __shfl_xor(s, off, 32);
    if (lane == 0) red[wave] = s;
    __syncthreads();
    if (tid < 32) {
        float v = red[tid];
        for (int off = 16; off > 0; off >>= 1) v += __shfl_xor(v, off, 32);
        if (tid == 0) red[0] = v;
    }
    __syncthreads();
    const float lse = gmax + logf(red[0]);

    // pass 3: write (each thread only touches its own disjoint stride set)
    for (int i = tid; i < V; i += 1024) logp[i] = logp[i] - lse;
}

// ---------------------------------------------------------------------------
// Launcher
// ---------------------------------------------------------------------------
extern "C" void kernel_launch(void* const* d_in, const int* in_sizes, int n_in,
                              void* d_out, int out_size, void* d_ws, size_t ws_size,
                              hipStream_t stream) {
    const int*   input = (const int*)  d_in[0];
    const float* h0    = (const float*)d_in[1];
    const float* c0    = (const float*)d_in[2];
    const float* emb   = (const float*)d_in[3];
    const float* Wih   = (const float*)d_in[4];
    const float* Whh   = (const float*)d_in[5];
    const float* bih   = (const float*)d_in[6];
    const float* bhh   = (const float*)d_in[7];
    const float* Wout  = (const float*)d_in[8];
    const float* bout  = (const float*)d_in[9];

    float* out    = (float*)d_out;         // [0,V): logp | [V,V+H): h_new | [V+H,V+2H): c_new
    float* ws     = (float*)d_ws;
    float* x      = ws;                    // H floats
    float* gates  = ws + H;                // 4H floats
    float* logits = out;                   // in-place with logp
    float* hnew   = out + V;

    k_embed_relu<<<(H + 255) / 256, 256, 0, stream>>>(input, emb, x);
    k_gates<<<(4 * H) / 16 / 8, 256, 0, stream>>>(x, h0, Wih, Whh, bih, bhh, gates);
    k_lstm<<<(H + 255) / 256, 256, 0, stream>>>(gates, c0, out);
    const int nStrips = (V + 15) / 16;
    k_logits<<<(nStrips + 7) / 8, 256, 0, stream>>>(hnew, Wout, bout, logits);
    k_logsoftmax<<<1, 1024, 0, stream>>>(logits);
}